// MonteCarloLRF_15247133900969
// MI455X (gfx1250) — compile-verified
//
#include <hip/hip_runtime.h>

#define B_ 4
#define N_ 4096
#define P_ 32
#define Q_ 32
#define K_ 8
#define TI 32

typedef __attribute__((ext_vector_type(4))) unsigned u32x4;
typedef __attribute__((ext_vector_type(8))) unsigned u32x8;
typedef __attribute__((ext_vector_type(2))) float    v2f;
typedef __attribute__((ext_vector_type(8))) float    v8f;

__global__ __launch_bounds__(1024) void MonteCarloLRF_gather_kernel(
    const float* __restrict__ x,     // (B, N, P)
    const int*   __restrict__ idx,   // (N, P, Q, K)
    const float* __restrict__ kw,    // (K, P, Q)
    const float* __restrict__ bias,  // (Q,)
    float*       __restrict__ out)   // (B, N, Q)
{
  // Double-buffered x column: xcol[buf][n][b] -> 2 * 4096 * 4 * 4B = 128 KB
  __shared__ float xcol[2][N_][B_];
  // Kernel weights, original [j][c][f] layout -> 32 KB
  __shared__ float kwl[K_ * P_ * Q_];

  const int f   = threadIdx.x;          // 0..31  (Q)
  const int iy  = threadIdx.y;          // 0..31  (i within tile)
  const int tid = iy * 32 + f;          // 0..1023
  const int i   = blockIdx.x * TI + iy; // row index in N

  // ---- TDM: stage the 8192-float contiguous weight tensor into LDS ----
  // Wave-uniform guard: only wave 0 (iy==0) issues the TDM op (EXEC ignored by TDM).
  if (iy == 0) {
    unsigned long long ga = (unsigned long long)(const void*)kw;
    unsigned ldsa = (unsigned)(size_t)(void*)&kwl[0];
    u32x4 g0;
    g0.x = 1u;                                                  // count=1 (valid D#)
    g0.y = ldsa;                                                // lds_addr (bytes)
    g0.z = (unsigned)ga;                                        // global_addr[31:0]
    g0.w = ((unsigned)(ga >> 32) & 0x01FFFFFFu) | (2u << 30);   // addr[56:32] | type=2
    u32x8 g1;
    g1.s0 = 2u << 16;        // workgroup_mask=0 (not in cluster), data_size=2 (4B)
    g1.s1 = 8192u << 16;     // tensor_dim0 = 8192 (lo16 at bits 63:48)
    g1.s2 = 1u << 16;        // tensor_dim0 hi16 = 0 | tensor_dim1 = 1
    g1.s3 = 8192u << 16;     // tensor_dim1 hi16 = 0 | tile_dim0 = 8192
    g1.s4 = 1u;              // tile_dim1 = 1, tile_dim2 = 0
    g1.s5 = 8192u;           // tensor_dim0_stride lo32
    g1.s6 = 0u;              // stride0 hi16 | stride1 lo16
    g1.s7 = 0u;              // stride1 hi32
    asm volatile("tensor_load_to_lds %0, %1" :: "s"(g0), "s"(g1) : "memory");
    __builtin_amdgcn_s_wait_tensorcnt(0);
  }

  const unsigned xbase = (unsigned)(size_t)(void*)&xcol[0][0][0];

  // Stage x[:, :, c] (16384 floats) into xcol[buf][n][b] via async global->LDS.
  // 16 wave-wide async b32 ops per wave per round (ASYNCcnt += 16).
  auto stage = [&](int c, int buf) {
    const unsigned bufofs = xbase + (unsigned)(buf << 16); // 64 KB per buffer
#pragma unroll
    for (int k = 0; k < 16; ++k) {
      int e = tid + (k << 10);           // 0..16383
      int b = e >> 12;                   // batch
      int n = e & (N_ - 1);              // row
      unsigned lofs = bufofs + (unsigned)(n * 16 + b * 4);
      unsigned long long g =
          (unsigned long long)(const void*)(x + ((size_t)b << 17) + n * 32 + c);
      asm volatile("global_load_async_to_lds_b32 %0, %1, off"
                   :: "v"(lofs), "v"(g) : "memory");
    }
  };

  stage(0, 0);

  float ax = 0.f, ay = 0.f, az = 0.f, aw = 0.f;
  // idx[i][c][f][j] flat = i*8192 + c*256 + f*8 ; base at c=0, as int4 (16B aligned)
  const int4* ip = (const int4*)(idx + (size_t)i * 8192 + (size_t)f * 8);

  for (int c = 0; c < P_; ++c) {
    if (c + 1 < P_) {
      stage(c + 1, (c + 1) & 1);
      // 16 new ops outstanding; waiting to <=16 guarantees round c completed
      asm volatile("s_wait_asynccnt 0x10" ::: "memory");
    } else {
      asm volatile("s_wait_asynccnt 0x0" ::: "memory");
    }
    __syncthreads();  // all waves' round-c LDS writes (and TDM at c==0) visible

    const int4* ipc = ip + (size_t)c * 64;  // +256 ints per c
    if (c + 2 < P_) __builtin_prefetch((const char*)ipc + 2048, 0, 1);
    const int4 q0 = ipc[0];
    const int4 q1 = ipc[1];
    const float* xb = &xcol[c & 1][0][0];
    const int nn[8] = {q0.x, q0.y, q0.z, q0.w, q1.x, q1.y, q1.z, q1.w};
#pragma unroll
    for (int j = 0; j < 8; ++j) {
      // one ds_load_b128 fetches x[0..3][n] for this gathered index
      const float4 v = *(const float4*)(xb + (size_t)(unsigned)nn[j] * 4);
      const float  w = kwl[(j * 32 + c) * 32 + f];  // conflict-free: lanes span f
      ax = fmaf(v.x, w, ax);
      ay = fmaf(v.y, w, ay);
      az = fmaf(v.z, w, az);
      aw = fmaf(v.w, w, aw);
    }
    __syncthreads();  // everyone done reading buf[c&1] before it is restaged
  }

  // ---- Epilogue: fused bias broadcast-add on the matrix unit ----
  // D = A x B + C with 16x16x4 f32 WMMA.
  //   A (16x4): A[m,0]=1 for m<8, A[m,1]=1 for 8<=m<16  (selector rows)
  //     lane layout: lanes 0-15 hold K0/K1 (a0,a1), lanes 16-31 hold K2/K3 (=0)
  //   B (4x16): B[0,n]=bias[n], B[1,n]=bias[n+16] -> per-lane b0 = bias[f], b1 = 0
  //   C: VGPR r = acc of batch r (rows M=r for f<16, M=r+8 for f>=16); rows 4..7 = 0
  // => D[r] at lane f == acc[b=r][f] + bias[f] for all 32 lanes (exact, any bias).
  // (The main contraction cannot use WMMA: the gathered operand depends on the
  //  output column f through idx[i,c,f,j], so no shared-operand matmul exists.)
  {
    const float bb = bias[f];
    v2f A;
    A.x = (f < 8) ? 1.0f : 0.0f;
    A.y = (f >= 8 && f < 16) ? 1.0f : 0.0f;
    v2f Bv;
    Bv.x = bb;
    Bv.y = 0.0f;
    v8f C;
    C[0] = ax; C[1] = ay; C[2] = az; C[3] = aw;
    C[4] = 0.f; C[5] = 0.f; C[6] = 0.f; C[7] = 0.f;
    v8f D = __builtin_amdgcn_wmma_f32_16x16x4_f32(
        false, A, false, Bv, (short)0, C, false, false);

    const int o = i * 32 + f;               // (i, f) within one batch plane
    out[o]                = D[0];
    out[o + N_ * Q_]      = D[1];
    out[o + 2 * N_ * Q_]  = D[2];
    out[o + 3 * N_ * Q_]  = D[3];
  }
}

extern "C" void kernel_launch(void* const* d_in, const int* in_sizes, int n_in,
                              void* d_out, int out_size, void* d_ws, size_t ws_size,
                              hipStream_t stream) {
  const float* x    = (const float*)d_in[0];
  const int*   idx  = (const int*)d_in[1];
  const float* kern = (const float*)d_in[2];
  const float* bias = (const float*)d_in[3];
  float*       out  = (float*)d_out;
  (void)in_sizes; (void)n_in; (void)out_size; (void)d_ws; (void)ws_size;

  dim3 grid(N_ / TI);     // 128 workgroups
  dim3 block(Q_, TI);     // (32, 32) = 1024 threads = 32 waves
  MonteCarloLRF_gather_kernel<<<grid, block, 0, stream>>>(x, idx, kern, bias, out);
}